// SimpleVolumeRenderer_65060164600145
// MI455X (gfx1250) — compile-verified
//
#include <hip/hip_runtime.h>
#include <stdint.h>

// ---------------------------------------------------------------------------
// NeRF volume rendering: one wave32 per ray, async global->LDS double buffer.
//   N_RAYS = 65536, N_SAMPLES = 128 (4 chunks of 32 samples per wave).
// Outputs (concatenated in d_out): rgb_map[N,3], depth_map[N], weight[N,128],
//   masked_weights[N,128], acc_map[N], disp_map[N].
// ---------------------------------------------------------------------------

#define EPSILON_F 1.0e10f
#define ATTEN_THR 0.001f
#define NSAMP 128
#define WAVES_PER_BLOCK 8
#define RPW 4 // rays per wave (pipelined)

// Bytes per ray buffer in LDS: 2048 (radiance 128*4 f32) + 512 (depth 128 f32)
#define RAY_BUF_BYTES 2560
#define RAY_BUF_FLOATS 640

// --- CDNA5 async global->LDS DMA (ASYNCcnt-tracked), GVS addressing ---------
// lds   : byte address within workgroup LDS allocation (VDST vgpr)
// voff  : per-lane 32-bit byte offset (VADDR vgpr)
// sbase : uniform 64-bit global base (SADDR sgpr pair)
// IO    : instruction offset, added to BOTH lds and global addresses
template <int IO>
__device__ __forceinline__ void async_copy_b128(uint32_t lds, uint32_t voff,
                                                uint64_t sbase) {
  asm volatile("global_load_async_to_lds_b128 %0, %1, %2 offset:%3"
               :
               : "v"(lds), "v"(voff), "s"(sbase), "n"(IO)
               : "memory");
}
__device__ __forceinline__ void wait_async_le(int has_next) {
  if (has_next)
    asm volatile("s_wait_asynccnt 5" ::: "memory"); // prev 5 done, next 5 fly
  else
    asm volatile("s_wait_asynccnt 0" ::: "memory");
}
__device__ __forceinline__ void wait_ds0() {
  asm volatile("s_wait_dscnt 0" ::: "memory"); // fence LDS reads before reuse
}

__global__ __launch_bounds__(256) void nerf_render_kernel(
    const float* __restrict__ rf,     // [N,128,4]
    const float* __restrict__ depth,  // [N,128]
    const float* __restrict__ dirs,   // [N,3]
    float* __restrict__ rgb_map, float* __restrict__ depth_map,
    float* __restrict__ weight, float* __restrict__ masked,
    float* __restrict__ acc_map, float* __restrict__ disp_map, int n_rays) {
  __shared__ __align__(16) float smem[WAVES_PER_BLOCK][2][RAY_BUF_FLOATS];

  const int lane = threadIdx.x & 31;
  const int wid = threadIdx.x >> 5;
  const int wave = blockIdx.x * WAVES_PER_BLOCK + wid;
  const int ray0 = wave * RPW;
  const uint32_t lvoff = (uint32_t)lane * 16u;
  // smem is the only __shared__ object -> starts at LDS offset 0.
  const uint32_t ldsWave = (uint32_t)(wid * 2 * RAY_BUF_BYTES);

  auto issue = [&](int buf, int ray) {
    const uint64_t rbase = (uint64_t)(uintptr_t)(rf + (size_t)ray * (NSAMP * 4));
    const uint64_t dbase = (uint64_t)(uintptr_t)(depth + (size_t)ray * NSAMP);
    const uint32_t l = ldsWave + (uint32_t)buf * RAY_BUF_BYTES + lvoff;
    async_copy_b128<0>(l, lvoff, rbase);      // radiance bytes [0,512)
    async_copy_b128<512>(l, lvoff, rbase);    // [512,1024)
    async_copy_b128<1024>(l, lvoff, rbase);   // [1024,1536)
    async_copy_b128<1536>(l, lvoff, rbase);   // [1536,2048)
    async_copy_b128<0>(l + 2048u, lvoff, dbase); // depth 512B
  };

  if (ray0 < n_rays) issue(0, ray0); // prime the pipeline

#pragma unroll
  for (int i = 0; i < RPW; ++i) {
    const int ray = ray0 + i;
    const bool cur_ok = (ray < n_rays);
    const bool next_ok = (i + 1 < RPW) && (ray + 1 < n_rays);
    if (!cur_ok) break;
    if (next_ok) {
      wait_ds0();               // prior reads of the reused buffer retired
      issue((i + 1) & 1, ray + 1);
    }
    wait_async_le(next_ok ? 1 : 0); // current buffer's 5 DMAs complete

    const int buf = i & 1;
    const float4* rfL = (const float4*)&smem[wid][buf][0];
    const float* dL = &smem[wid][buf][NSAMP * 4];

    // ||ray_direction|| (uniform per wave)
    const float dx = dirs[(size_t)ray * 3 + 0];
    const float dy = dirs[(size_t)ray * 3 + 1];
    const float dz = dirs[(size_t)ray * 3 + 2];
    const float nrm = sqrtf(dx * dx + dy * dy + dz * dz);

    float carry = 1.0f; // product of t over all previous chunks
    float sr = 0.f, sg = 0.f, sb = 0.f, sacc = 0.f, sdep = 0.f;

#pragma unroll
    for (int k = 0; k < 4; ++k) {
      const int s = k * 32 + lane;
      const float4 c = rfL[s];     // ds_load_b128 (rgb + sigma)
      const float d = dL[s];       // ds_load_b32
      const int sn = (s < NSAMP - 1) ? (s + 1) : (NSAMP - 1);
      const float dn = dL[sn];
      const float delta =
          ((s < NSAMP - 1) ? (dn - d) : EPSILON_F) * nrm;

      const float sigma = fmaxf(c.w, 0.0f);
      const float e = expf(-sigma * delta);
      const float alpha = 1.0f - e;
      const float t = (1.0f - alpha) + 1e-10f;

      // wave32 inclusive product scan over this 32-sample chunk
      float v = t;
#pragma unroll
      for (int off = 1; off < 32; off <<= 1) {
        const float u = __shfl_up(v, off, 32);
        if (lane >= off) v *= u;
      }
      float excl = __shfl_up(v, 1, 32);
      if (lane == 0) excl = 1.0f;
      const float T = carry * excl;          // exclusive transmittance
      carry *= __shfl(v, 31, 32);            // chunk total -> carry

      const float w = alpha * T;
      weight[(size_t)ray * NSAMP + s] = w;
      masked[(size_t)ray * NSAMP + s] = (T > ATTEN_THR) ? 1.0f : 0.0f;

      sr += w * c.x;
      sg += w * c.y;
      sb += w * c.z;
      sacc += w;
      sdep += w * d;
    }

    // wave reductions (butterfly)
#pragma unroll
    for (int off = 16; off > 0; off >>= 1) {
      sr += __shfl_xor(sr, off, 32);
      sg += __shfl_xor(sg, off, 32);
      sb += __shfl_xor(sb, off, 32);
      sacc += __shfl_xor(sacc, off, 32);
      sdep += __shfl_xor(sdep, off, 32);
    }

    if (lane == 0) {
      rgb_map[(size_t)ray * 3 + 0] = sr;
      rgb_map[(size_t)ray * 3 + 1] = sg;
      rgb_map[(size_t)ray * 3 + 2] = sb;
      depth_map[ray] = sdep;
      acc_map[ray] = sacc;
      const float ratio = sdep / sacc;             // 0/0 -> NaN
      float disp = 1.0f / fmaxf(1e-10f, ratio);
      if (ratio != ratio) disp = 0.0f;             // jnp: NaN -> 0
      disp_map[ray] = disp;
    }
  }
}

extern "C" void kernel_launch(void* const* d_in, const int* in_sizes, int n_in,
                              void* d_out, int out_size, void* d_ws,
                              size_t ws_size, hipStream_t stream) {
  const float* rf = (const float*)d_in[0];    // [N,128,4]
  const float* depth = (const float*)d_in[1]; // [N,128]
  const float* dirs = (const float*)d_in[2];  // [N,3]
  const int N = in_sizes[2] / 3;

  float* out = (float*)d_out;
  float* rgb_map = out;                                  // [N,3]
  float* depth_map = out + (size_t)3 * N;                // [N]
  float* weight = out + (size_t)4 * N;                   // [N,128]
  float* masked = weight + (size_t)N * NSAMP;            // [N,128]
  float* acc_map = masked + (size_t)N * NSAMP;           // [N]
  float* disp_map = acc_map + N;                         // [N]

  const int waves = (N + RPW - 1) / RPW;
  const int blocks = (waves + WAVES_PER_BLOCK - 1) / WAVES_PER_BLOCK;
  nerf_render_kernel<<<blocks, WAVES_PER_BLOCK * 32, 0, stream>>>(
      rf, depth, dirs, rgb_map, depth_map, weight, masked, acc_map, disp_map,
      N);
}